// PyG_model_54863912239540
// MI455X (gfx1250) — compile-verified
//
#include <hip/hip_runtime.h>
#include <hip/hip_bf16.h>

// N = 50000, E = 800000, D = H = 64, O = 32, G = 64
// Inputs: 0:x[N,64] 1:edge_index[2,E] 2:batch[N] 3:W1[64,64] 4:b1[64]
//         5:W2 6:b2 7:W3 8:b3 9:Wl[64,32] 10:bl[32]
// Output: [G, O] = [64, 32] fp32

typedef __attribute__((ext_vector_type(2))) float v2f;
typedef __attribute__((ext_vector_type(8))) float v8f;

#define HH 64   // hidden dim
#define OO 32
#define GG 64

// ---------------- degree / normalization ----------------

__global__ void gcn_deg_init(float* __restrict__ deg, int n) {
    int i = blockIdx.x * blockDim.x + threadIdx.x;
    if (i < n) deg[i] = 1.0f;                       // self-loop contributes 1
}

__global__ void gcn_deg_edges(const int* __restrict__ ei, float* __restrict__ deg,
                              int E_) {
    int e = blockIdx.x * blockDim.x + threadIdx.x;
    if (e < E_) atomicAdd(&deg[ei[E_ + e]], 1.0f);  // dst = edge_index[1][e]
}

__global__ void gcn_rsqrt(float* __restrict__ deg, int n) {
    int i = blockIdx.x * blockDim.x + threadIdx.x;
    if (i < n) deg[i] = rsqrtf(deg[i]);             // in-place: deg -> dis
}

// ---------------- fp32 WMMA GEMM: HW = f(H) @ W ----------------
// f(H) = relu(H + bias) when applyReluBias, else H.
// One wave computes a 16x64 output tile using V_WMMA_F32_16X16X4_F32.
// A 16x4 f32 layout (ISA 7.12.2): lane l holds row (l&15), K = step*4 + (l>>4)*2 {+0,+1}.
// C/D 16x16 f32 layout: VGPR j, lanes 0-15 -> M=j, lanes 16-31 -> M=j+8; N = lane&15.
//
// W is staged once per block into LDS, pre-swizzled into per-lane fragment
// order: bfrag[s*4+ct][lane] is the float2 B-operand for K-step s, col-tile ct.
// Inner loop B reads are then single ds_load_b64 at immediate offsets (low
// latency) instead of two strided global loads with a full loadcnt-0 stall
// before every WMMA. 4 independent accumulator chains interleaved for ILP.

__global__ __launch_bounds__(128) void gcn_gemm_wmma(
    const float* __restrict__ Hm,     // [n, 64]
    const float* __restrict__ W,      // [64, 64] row-major
    const float* __restrict__ bias,   // [64] (may be null when !applyReluBias)
    float* __restrict__ HW,           // [n, 64]
    int n, int applyReluBias)
{
    __shared__ v2f bfrag[64 * 32];                  // 16 KB: [frag f][lane]

    const int tid = threadIdx.x;

    // Stage + swizzle W into LDS (all 128 threads, before any wave exits).
#pragma unroll
    for (int i = tid; i < 64 * 32; i += 128) {
        const int f  = i >> 5;                      // fragment id = s*4 + ct
        const int s  = f >> 2;
        const int ct = f & 3;
        const int ln = i & 31;
        const int hs = ln >> 4;
        const int ld = ln & 15;
        const int k  = s * 4 + hs * 2;
        v2f b;
        b.x = W[(size_t)(k    ) * HH + ct * 16 + ld];
        b.y = W[(size_t)(k + 1) * HH + ct * 16 + ld];
        bfrag[i] = b;
    }
    __syncthreads();

    const int lane    = tid & 31;
    const int wave    = tid >> 5;
    const int rowBase = (blockIdx.x * 4 + wave) * 16;
    if (rowBase >= n) return;                       // uniform per-wave

    const int halfSel = lane >> 4;                  // 0 or 1
    const int lid     = lane & 15;
    const int row     = min(rowBase + lid, n - 1);  // clamp: no EXEC divergence

    // Preload A fragments for all 16 K-steps (32 VGPRs).
    v2f a[16];
#pragma unroll
    for (int s = 0; s < 16; ++s) {
        const int k = s * 4 + halfSel * 2;
        float x0 = Hm[(size_t)row * HH + k];
        float x1 = Hm[(size_t)row * HH + k + 1];
        if (applyReluBias) {
            x0 = fmaxf(x0 + bias[k], 0.0f);
            x1 = fmaxf(x1 + bias[k + 1], 0.0f);
        }
        a[s].x = x0;
        a[s].y = x1;
    }

    // 4 independent accumulator chains, B from LDS, interleaved across K-steps.
    v8f c0 = {}, c1 = {}, c2 = {}, c3 = {};
#pragma unroll
    for (int s = 0; s < 16; ++s) {
        const v2f b0 = bfrag[(s * 4 + 0) * 32 + lane];
        const v2f b1 = bfrag[(s * 4 + 1) * 32 + lane];
        const v2f b2 = bfrag[(s * 4 + 2) * 32 + lane];
        const v2f b3 = bfrag[(s * 4 + 3) * 32 + lane];
        c0 = __builtin_amdgcn_wmma_f32_16x16x4_f32(false, a[s], false, b0, (short)0, c0, false, false);
        c1 = __builtin_amdgcn_wmma_f32_16x16x4_f32(false, a[s], false, b1, (short)0, c1, false, false);
        c2 = __builtin_amdgcn_wmma_f32_16x16x4_f32(false, a[s], false, b2, (short)0, c2, false, false);
        c3 = __builtin_amdgcn_wmma_f32_16x16x4_f32(false, a[s], false, b3, (short)0, c3, false, false);
    }

    if (rowBase + 16 <= n) {
        // Fast path: unguarded, branch-free stores (always taken: N % 16 == 0).
#pragma unroll
        for (int j = 0; j < 8; ++j) {
            const size_t r = (size_t)(rowBase + j + 8 * halfSel) * HH + lid;
            HW[r +  0] = c0[j];
            HW[r + 16] = c1[j];
            HW[r + 32] = c2[j];
            HW[r + 48] = c3[j];
        }
    } else {
#pragma unroll
        for (int j = 0; j < 8; ++j) {
            const int rr = rowBase + j + 8 * halfSel;
            if (rr < n) {
                const size_t r = (size_t)rr * HH + lid;
                HW[r +  0] = c0[j];
                HW[r + 16] = c1[j];
                HW[r + 32] = c2[j];
                HW[r + 48] = c3[j];
            }
        }
    }
}

// ---------------- self-loop init: agg[i] = dis[i]^2 * hw[i] ----------------

__global__ __launch_bounds__(256) void gcn_selfloop_init(
    const float* __restrict__ hw, const float* __restrict__ dis,
    float* __restrict__ agg, int n)
{
    int idx = blockIdx.x * blockDim.x + threadIdx.x;  // one float2 each
    int total = n * (HH / 2);
    if (idx >= total) return;
    int node = idx / (HH / 2);
    float d = dis[node];
    float s = d * d;
    float2 v = ((const float2*)hw)[idx];
    float2 r; r.x = v.x * s; r.y = v.y * s;
    ((float2*)agg)[idx] = r;
}

// ---------------- edge scatter: agg[dst] += dis[src]*dis[dst] * hw[src] ----------------
// One wave32 per edge; lane handles 2 consecutive floats of the 64-float row.

__global__ __launch_bounds__(256) void gcn_edge_scatter(
    const int* __restrict__ ei, const float* __restrict__ hw,
    const float* __restrict__ dis, float* __restrict__ agg, int E_)
{
    int e = blockIdx.x * 8 + (threadIdx.x >> 5);
    if (e >= E_) return;                             // uniform per-wave
    const int lane = threadIdx.x & 31;
    const int src = ei[e];
    const int dst = ei[E_ + e];
    const float nrm = dis[src] * dis[dst];
    float2 v = ((const float2*)(hw + (size_t)src * HH))[lane];
    float* d = agg + (size_t)dst * HH + lane * 2;
    atomicAdd(d,     v.x * nrm);
    atomicAdd(d + 1, v.y * nrm);
}

// ---------------- pooling ----------------

__global__ void gcn_pool_zero(float* __restrict__ pooled, float* __restrict__ cnt) {
    int i = blockIdx.x * blockDim.x + threadIdx.x;
    if (i < GG * HH) pooled[i] = 0.0f;
    if (i < GG) cnt[i] = 0.0f;
}

__global__ __launch_bounds__(256) void gcn_pool_sum(
    const float* __restrict__ agg, const float* __restrict__ b3,
    const int* __restrict__ batch, float* __restrict__ pooled,
    float* __restrict__ cnt, int n)
{
    int node = blockIdx.x * 8 + (threadIdx.x >> 5);
    if (node >= n) return;
    const int lane = threadIdx.x & 31;
    const int g = batch[node];
    float2 v = ((const float2*)(agg + (size_t)node * HH))[lane];
    float* p = pooled + (size_t)g * HH + lane * 2;
    atomicAdd(p,     v.x + b3[lane * 2]);
    atomicAdd(p + 1, v.y + b3[lane * 2 + 1]);
    if (lane == 0) atomicAdd(&cnt[g], 1.0f);
}

// ---------------- final head: out = (pooled/cnt) @ Wl + bl ----------------

__global__ void gcn_head(const float* __restrict__ pooled, const float* __restrict__ cnt,
                         const float* __restrict__ Wl, const float* __restrict__ bl,
                         float* __restrict__ out)
{
    int idx = blockIdx.x * blockDim.x + threadIdx.x;
    if (idx >= GG * OO) return;
    const int g = idx / OO;
    const int o = idx % OO;
    const float inv = 1.0f / cnt[g];
    float s = bl[o];
#pragma unroll
    for (int h = 0; h < HH; ++h)
        s += pooled[(size_t)g * HH + h] * inv * Wl[(size_t)h * OO + o];
    out[idx] = s;
}

// ---------------- launcher ----------------

extern "C" void kernel_launch(void* const* d_in, const int* in_sizes, int n_in,
                              void* d_out, int out_size, void* d_ws, size_t ws_size,
                              hipStream_t stream) {
    const float* x    = (const float*)d_in[0];
    const int*   ei   = (const int*)  d_in[1];
    const int*   bat  = (const int*)  d_in[2];
    const float* W1   = (const float*)d_in[3];
    const float* b1   = (const float*)d_in[4];
    const float* W2   = (const float*)d_in[5];
    const float* b2   = (const float*)d_in[6];
    const float* W3   = (const float*)d_in[7];
    const float* b3   = (const float*)d_in[8];
    const float* Wl   = (const float*)d_in[9];
    const float* bl   = (const float*)d_in[10];
    float*       out  = (float*)d_out;

    const int N = in_sizes[0] / HH;    // 50000
    const int E = in_sizes[1] / 2;     // 800000

    // Workspace layout (floats)
    float* ws  = (float*)d_ws;
    size_t off = 0;
    float* dis    = ws + off; off += (size_t)((N + 15) & ~15);
    float* hw     = ws + off; off += (size_t)N * HH;
    float* agg    = ws + off; off += (size_t)N * HH;
    float* pooled = ws + off; off += GG * HH;
    float* cnt    = ws + off; off += GG;

    const int TPB = 256;
    const int nBlkN   = (N + TPB - 1) / TPB;
    const int nBlkE   = (E + TPB - 1) / TPB;
    const int nTiles  = (N + 15) / 16;
    const int gemmBlk = (nTiles + 3) / 4;               // 4 waves/block
    const int slBlk   = (N * (HH / 2) + TPB - 1) / TPB; // self-loop float2s
    const int edgBlk  = (E + 7) / 8;                    // 8 waves/block, 1 edge/wave
    const int poolBlk = (N + 7) / 8;

    // Degrees -> dis = rsqrt(deg)
    gcn_deg_init <<<nBlkN, TPB, 0, stream>>>(dis, N);
    gcn_deg_edges<<<nBlkE, TPB, 0, stream>>>(ei, dis, E);
    gcn_rsqrt    <<<nBlkN, TPB, 0, stream>>>(dis, N);

    // Layer 1: hw = x @ W1 ; agg = selfloop + scatter
    gcn_gemm_wmma    <<<gemmBlk, 128, 0, stream>>>(x,   W1, nullptr, hw, N, 0);
    gcn_selfloop_init<<<slBlk,   TPB, 0, stream>>>(hw, dis, agg, N);
    gcn_edge_scatter <<<edgBlk,  TPB, 0, stream>>>(ei, hw, dis, agg, E);

    // Layer 2: hw = relu(agg + b1) @ W2 ; agg = selfloop + scatter
    gcn_gemm_wmma    <<<gemmBlk, 128, 0, stream>>>(agg, W2, b1, hw, N, 1);
    gcn_selfloop_init<<<slBlk,   TPB, 0, stream>>>(hw, dis, agg, N);
    gcn_edge_scatter <<<edgBlk,  TPB, 0, stream>>>(ei, hw, dis, agg, E);

    // Layer 3: hw = relu(agg + b2) @ W3 ; agg = selfloop + scatter (b3 fused in pool)
    gcn_gemm_wmma    <<<gemmBlk, 128, 0, stream>>>(agg, W3, b2, hw, N, 1);
    gcn_selfloop_init<<<slBlk,   TPB, 0, stream>>>(hw, dis, agg, N);
    gcn_edge_scatter <<<edgBlk,  TPB, 0, stream>>>(ei, hw, dis, agg, E);

    // Mean pool + head
    gcn_pool_zero<<<(GG * HH + TPB - 1) / TPB, TPB, 0, stream>>>(pooled, cnt);
    gcn_pool_sum <<<poolBlk, TPB, 0, stream>>>(agg, b3, bat, pooled, cnt, N);
    gcn_head     <<<(GG * OO + TPB - 1) / TPB, TPB, 0, stream>>>(pooled, cnt, Wl, bl, out);
}